// GCN_76416058131283
// MI455X (gfx1250) — compile-verified
//
#include <hip/hip_runtime.h>
#include <hip/hip_bf16.h>

#define NODES  100000
#define NEDGE  1600000
#define NGRAPH 64
#define HID    128

typedef float v2f __attribute__((ext_vector_type(2)));
typedef float v8f __attribute__((ext_vector_type(8)));
typedef int   gv4i __attribute__((vector_size(4 * sizeof(int))));   // matches builtin param

// Async Global->LDS staging (gfx1250 ASYNCcnt path) if the toolchain has it.
#if defined(__has_builtin)
#  if __has_builtin(__builtin_amdgcn_global_load_async_to_lds_b128) && \
      __has_builtin(__builtin_amdgcn_s_wait_asynccnt)
#    define USE_ASYNC_LDS 1
#  endif
#endif
#ifndef USE_ASYNC_LDS
#  define USE_ASYNC_LDS 0
#endif

// ---------------------------------------------------------------------------
// Degree / normalization
// ---------------------------------------------------------------------------
__global__ void deg_init_kernel(float* deg, int n) {
    int i = blockIdx.x * blockDim.x + threadIdx.x;
    if (i < n) deg[i] = 1.0f;              // self loop contributes 1
}

__global__ void deg_count_kernel(const int* __restrict__ dst, float* deg, int ne) {
    int e = blockIdx.x * blockDim.x + threadIdx.x;
    if (e < ne) atomicAdd(&deg[dst[e]], 1.0f);
}

__global__ void deg_to_dinv_kernel(float* deg, int n) {
    int i = blockIdx.x * blockDim.x + threadIdx.x;
    if (i < n) deg[i] = rsqrtf(fmaxf(deg[i], 1e-12f));
}

// ---------------------------------------------------------------------------
// Layer-1 GEMM: T = x(N x 6) @ W1(6 x 128)   (K too small for WMMA)
// ---------------------------------------------------------------------------
__global__ void gemm_in6_kernel(const float* __restrict__ x,
                                const float* __restrict__ W1,
                                float* __restrict__ T, int n) {
    int idx = blockIdx.x * blockDim.x + threadIdx.x;
    if (idx >= n * HID) return;
    int node = idx >> 7;
    int f    = idx & (HID - 1);
    const float* xr = x + node * 6;
    float acc = 0.f;
#pragma unroll
    for (int k = 0; k < 6; ++k) acc += xr[k] * W1[k * HID + f];
    T[idx] = acc;
}

// ---------------------------------------------------------------------------
// WMMA GEMM: T = H(nrows x 128) @ W(128 x 128), fp32 via V_WMMA_F32_16X16X4_F32
// One wave per 16x16 output tile; 8 waves/block cover all 128 columns of a
// 16-row strip. W staged into LDS via async Global->LDS B128 copies (ASYNCcnt)
// when available. All pre-WMMA control flow is wave-uniform (EXEC all ones).
// ---------------------------------------------------------------------------
__global__ __launch_bounds__(256) void gemm128_wmma_kernel(
        const float* __restrict__ H, const float* __restrict__ W,
        float* __restrict__ T, int nrows) {
    __shared__ __align__(16) float sW[HID * HID];   // 64 KB

#if USE_ASYNC_LDS
    {
        char* gptr = (char*)W;
        char* lptr = (char*)sW;
        // 64 KB = 4096 x 16 B; 256 threads -> 16 async B128 issues each.
        for (int i = threadIdx.x * 16; i < HID * HID * 4; i += 256 * 16) {
            __builtin_amdgcn_global_load_async_to_lds_b128(
                (__attribute__((address_space(1))) gv4i*)(gptr + i),
                (__attribute__((address_space(3))) gv4i*)(lptr + i),
                0, 0);
        }
        __builtin_amdgcn_s_wait_asynccnt(0);
    }
    __syncthreads();
#else
    for (int i = threadIdx.x; i < HID * HID; i += 256) sW[i] = W[i];
    __syncthreads();
#endif

    const int wave  = threadIdx.x >> 5;   // 0..7 -> column tile
    const int lane  = threadIdx.x & 31;
    const int n0    = wave * 16;
    const int m0    = blockIdx.x * 16;    // uniform across wave
    if (m0 >= nrows) return;              // uniform branch

    const int mrow  = m0 + (lane & 15);   // A: lanes 0-15 / 16-31 both M=0..15
    const int khalf = (lane >> 4);        // 0 -> K=k,k+1 ; 1 -> K=k+2,k+3
    const int ncol  = n0 + (lane & 15);   // B/C/D: N striped over lanes

    v8f acc = {0.f, 0.f, 0.f, 0.f, 0.f, 0.f, 0.f, 0.f};
#pragma unroll 4
    for (int k = 0; k < HID; k += 4) {
        const int kb = k + khalf * 2;
        v2f a = *(const v2f*)(H + (size_t)mrow * HID + kb);      // contiguous K pair
        v2f b;
        b.x = sW[kb * HID + ncol];                               // B row K=kb
        b.y = sW[(kb + 1) * HID + ncol];                         // B row K=kb+1
        acc = __builtin_amdgcn_wmma_f32_16x16x4_f32(
                false, a, false, b, (short)0, acc, false, false);
    }
    // C/D layout: VGPR v holds row m0+v (lanes 0-15) or m0+v+8 (lanes 16-31)
#pragma unroll
    for (int v = 0; v < 8; ++v)
        T[(size_t)(m0 + v + 8 * khalf) * HID + ncol] = acc[v];
}

// ---------------------------------------------------------------------------
// Aggregation: A = T * dinv^2 (self loops), then scatter E edges, then
// H = relu(A + b).
// ---------------------------------------------------------------------------
__global__ void agg_init_kernel(const float* __restrict__ T,
                                const float* __restrict__ dinv,
                                float* __restrict__ A, int n) {
    int idx = blockIdx.x * blockDim.x + threadIdx.x;
    if (idx >= n * HID) return;
    float dv = dinv[idx >> 7];
    A[idx] = T[idx] * dv * dv;
}

__global__ __launch_bounds__(256) void scatter_kernel(
        const float* __restrict__ T, const float* __restrict__ dinv,
        const int* __restrict__ src, const int* __restrict__ dst,
        float* __restrict__ A, int ne) {
    int wid  = (blockIdx.x * blockDim.x + threadIdx.x) >> 5;   // one wave32/edge
    int lane = threadIdx.x & 31;
    if (wid >= ne) return;
    int s = src[wid];
    int d = dst[wid];
    float c = dinv[s] * dinv[d];
    const float* trow = T + (size_t)s * HID;
    float*       arow = A + (size_t)d * HID;
#pragma unroll
    for (int i = 0; i < 4; ++i) {
        int f = lane + i * 32;
        atomicAdd(&arow[f], trow[f] * c);
    }
}

__global__ void finalize_kernel(const float* __restrict__ A,
                                const float* __restrict__ b,
                                float* __restrict__ Hout, int n) {
    int idx = blockIdx.x * blockDim.x + threadIdx.x;
    if (idx >= n * HID) return;
    Hout[idx] = fmaxf(A[idx] + b[idx & (HID - 1)], 0.f);
}

// ---------------------------------------------------------------------------
// Global mean pool (batch is sorted: accumulate runs in registers, flush with
// rare atomics).
// ---------------------------------------------------------------------------
__global__ void zero_kernel(float* p, int n) {
    int i = blockIdx.x * blockDim.x + threadIdx.x;
    if (i < n) p[i] = 0.f;
}

__global__ __launch_bounds__(128) void pool_kernel(
        const float* __restrict__ H, const int* __restrict__ batch,
        float* __restrict__ psum, float* __restrict__ pcnt,
        int n, int nodes_per_block) {
    int f     = threadIdx.x;                       // feature 0..127
    int start = blockIdx.x * nodes_per_block;
    int end   = min(start + nodes_per_block, n);
    if (start >= end) return;
    float acc = 0.f, cnt = 0.f;
    int cur = batch[start];
    for (int i = start; i < end; ++i) {
        int g = batch[i];
        if (g != cur) {
            atomicAdd(&psum[cur * HID + f], acc);
            if (f == 0) atomicAdd(&pcnt[cur], cnt);
            acc = 0.f; cnt = 0.f; cur = g;
        }
        acc += H[(size_t)i * HID + f];
        cnt += 1.f;
    }
    atomicAdd(&psum[cur * HID + f], acc);
    if (f == 0) atomicAdd(&pcnt[cur], cnt);
}

__global__ void pmean_kernel(const float* __restrict__ psum,
                             const float* __restrict__ pcnt,
                             float* __restrict__ pmean, int n) {
    int idx = blockIdx.x * blockDim.x + threadIdx.x;
    if (idx >= n) return;
    pmean[idx] = psum[idx] / fmaxf(pcnt[idx >> 7], 1.0f);
}

// ---------------------------------------------------------------------------
// Head: relu(z + lin1_b) -> logits = z @ lin2_w + lin2_b -> softmax(2)
// (z = pmean @ lin1_w comes from the WMMA kernel)
// ---------------------------------------------------------------------------
__global__ __launch_bounds__(128) void head_kernel(
        const float* __restrict__ Z, const float* __restrict__ lin1_b,
        const float* __restrict__ lin2_w, const float* __restrict__ lin2_b,
        float* __restrict__ out) {
    __shared__ float zr[HID];
    __shared__ float logits[2];
    int g = blockIdx.x, f = threadIdx.x;
    zr[f] = fmaxf(Z[g * HID + f] + lin1_b[f], 0.f);
    __syncthreads();
    if (f < 2) {
        float acc = lin2_b[f];
        for (int k = 0; k < HID; ++k) acc += zr[k] * lin2_w[k * 2 + f];
        logits[f] = acc;
    }
    __syncthreads();
    if (f == 0) {
        float m  = fmaxf(logits[0], logits[1]);
        float e0 = __expf(logits[0] - m);
        float e1 = __expf(logits[1] - m);
        float s  = e0 + e1;
        out[g * 2 + 0] = e0 / s;
        out[g * 2 + 1] = e1 / s;
    }
}

// ---------------------------------------------------------------------------
// Launch
// ---------------------------------------------------------------------------
extern "C" void kernel_launch(void* const* d_in, const int* in_sizes, int n_in,
                              void* d_out, int out_size, void* d_ws, size_t ws_size,
                              hipStream_t stream) {
    const float* x      = (const float*)d_in[0];
    const int*   ei     = (const int*)  d_in[1];
    const int*   batch  = (const int*)  d_in[2];
    const float* W1     = (const float*)d_in[3];
    const float* b1     = (const float*)d_in[4];
    const float* Ws     = (const float*)d_in[5];   // 2 x 128 x 128
    const float* bs     = (const float*)d_in[6];   // 2 x 128
    const float* lin1_w = (const float*)d_in[7];
    const float* lin1_b = (const float*)d_in[8];
    const float* lin2_w = (const float*)d_in[9];
    const float* lin2_b = (const float*)d_in[10];
    float* out = (float*)d_out;

    const int* src = ei;
    const int* dst = ei + NEDGE;

    const size_t NH = (size_t)NODES * HID;
    float* buf0  = (float*)d_ws;          // activations
    float* buf1  = buf0 + NH;             // GEMM result T
    float* buf2  = buf1 + NH;             // aggregation accumulator
    float* dinv  = buf2 + NH;             // NODES
    float* psum  = dinv + NODES;          // G*128
    float* pcnt  = psum + NGRAPH * HID;   // G
    float* pmean = pcnt + NGRAPH;         // G*128
    float* zbuf  = pmean + NGRAPH * HID;  // G*128

    const int B256 = 256;
    const int gridNH   = (int)((NH + B256 - 1) / B256);
    const int gridN    = (NODES + B256 - 1) / B256;
    const int gridE    = (NEDGE + B256 - 1) / B256;
    const int gridEdgeWaves = (int)(((size_t)NEDGE * 32 + B256 - 1) / B256);

    // Degree -> dinv
    deg_init_kernel   <<<gridN, B256, 0, stream>>>(dinv, NODES);
    deg_count_kernel  <<<gridE, B256, 0, stream>>>(dst, dinv, NEDGE);
    deg_to_dinv_kernel<<<gridN, B256, 0, stream>>>(dinv, NODES);

    // Layer 1 (K=6, scalar GEMM)
    gemm_in6_kernel<<<gridNH, B256, 0, stream>>>(x, W1, buf1, NODES);
    agg_init_kernel<<<gridNH, B256, 0, stream>>>(buf1, dinv, buf2, NODES);
    scatter_kernel <<<gridEdgeWaves, B256, 0, stream>>>(buf1, dinv, src, dst, buf2, NEDGE);
    finalize_kernel<<<gridNH, B256, 0, stream>>>(buf2, b1, buf0, NODES);

    // Layers 2..3 (WMMA fp32 GEMMs)
    for (int l = 0; l < 2; ++l) {
        const float* W = Ws + (size_t)l * HID * HID;
        const float* b = bs + (size_t)l * HID;
        gemm128_wmma_kernel<<<NODES / 16, B256, 0, stream>>>(buf0, W, buf1, NODES);
        agg_init_kernel<<<gridNH, B256, 0, stream>>>(buf1, dinv, buf2, NODES);
        scatter_kernel <<<gridEdgeWaves, B256, 0, stream>>>(buf1, dinv, src, dst, buf2, NEDGE);
        finalize_kernel<<<gridNH, B256, 0, stream>>>(buf2, b, buf0, NODES);
    }

    // Global mean pool
    const int poolTot = NGRAPH * HID + NGRAPH;
    zero_kernel<<<(poolTot + B256 - 1) / B256, B256, 0, stream>>>(psum, poolTot); // psum+pcnt contiguous
    pool_kernel<<<(NODES + 127) / 128, 128, 0, stream>>>(buf0, batch, psum, pcnt, NODES, 128);
    pmean_kernel<<<(NGRAPH * HID + B256 - 1) / B256, B256, 0, stream>>>(psum, pcnt, pmean, NGRAPH * HID);

    // Head: z = pmean @ lin1_w via the same WMMA kernel (64 rows -> 4 blocks)
    gemm128_wmma_kernel<<<NGRAPH / 16, B256, 0, stream>>>(pmean, lin1_w, zbuf, NGRAPH);
    head_kernel<<<NGRAPH, 128, 0, stream>>>(zbuf, lin1_b, lin2_w, lin2_b, out);
}